// GraphAttention_85341000172247
// MI455X (gfx1250) — compile-verified
//
#include <hip/hip_runtime.h>
#include <hip/hip_bf16.h>

#define N_SPANS   4096
#define EMB_DIM   384
#define HEADS     4
#define PROJ_DIM  (HEADS * EMB_DIM)   // 1536
#define BAND      11                  // exp(-12/5) < 0.1 => |i-j| <= 11
#define BANDW     (2 * BAND + 1)      // 23
#define NEG_SLOPE 0.2f
#define THRESH    0.1f

typedef float v2f __attribute__((ext_vector_type(2)));
typedef float v8f __attribute__((ext_vector_type(8)));

__device__ __forceinline__ float wave_sum(float v) {
    #pragma unroll
    for (int o = 16; o > 0; o >>= 1) v += __shfl_xor(v, o, 32);
    return v;
}

// ---------------------------------------------------------------- kernel 1
// rnorm[i] = 1/||E[i]||  (one wave32 per row)
__global__ void __launch_bounds__(256) ga_norm_kernel(const float* __restrict__ E,
                                                      float* __restrict__ rnorm) {
    const int row  = blockIdx.x * 8 + (threadIdx.x >> 5);
    const int lane = threadIdx.x & 31;
    const float* e = E + row * EMB_DIM;
    float s = 0.f;
    #pragma unroll
    for (int k = lane; k < EMB_DIM; k += 32) { float v = e[k]; s += v * v; }
    s = wave_sum(s);
    if (lane == 0) rnorm[row] = rsqrtf(s);
}

// ---------------------------------------------------------------- kernel 2
// Banded adjacency: mask[i][jj] = (cos(e_i,e_j) * exp(-|i-j|/5) > 0.1), j = i-11+jj
__global__ void __launch_bounds__(256) ga_band_mask_kernel(const float* __restrict__ E,
                                                           const float* __restrict__ rnorm,
                                                           unsigned char* __restrict__ mask) {
    const int i    = blockIdx.x;
    const int wave = threadIdx.x >> 5;   // 0..7
    const int lane = threadIdx.x & 31;
    const float* ei = E + i * EMB_DIM;
    const float rni = rnorm[i];
    for (int jj = wave; jj < BANDW; jj += 8) {
        const int j = i - BAND + jj;
        unsigned char m = 0;
        if (j >= 0 && j < N_SPANS) {
            const float* ej = E + j * EMB_DIM;
            float s = 0.f;
            #pragma unroll
            for (int k = lane; k < EMB_DIM; k += 32) s += ei[k] * ej[k];
            s = wave_sum(s);
            const float d = fabsf((float)(i - j));
            const float a = s * rni * rnorm[j] * __expf(-d * 0.2f);
            m = (a > THRESH) ? 1 : 0;
        }
        if (lane == 0) mask[i * BANDW + jj] = m;
    }
}

// ---------------------------------------------------------------- kernel 3
// X[4096x1536] = E[4096x384] * Wf[384x1536], fp32 WMMA 16x16x4.
// One wave computes a 16(M) x 64(N) strip: A fragment reused across 4 B tiles.
// A layout (ISA 7.12.2, 32-bit 16x4): lane%16 = M, (lane/16)*2 = K base, 2 floats/lane.
// C/D layout: VGPR v -> M = v + (lane/16)*8, N = lane%16.
__global__ void __launch_bounds__(256) ga_gemm_wmma_kernel(const float* __restrict__ E,
                                                           const float* __restrict__ Wf,
                                                           float* __restrict__ X) {
    const int lane  = threadIdx.x & 31;
    const int wave  = threadIdx.x >> 5;          // 0..7
    const int mBase = (blockIdx.y * 8 + wave) * 16;
    const int nBase = blockIdx.x * 64;

    const int mRow = mBase + (lane & 15);
    const int kb   = (lane >> 4) << 1;           // 0 or 2
    const int nCol = nBase + (lane & 15);

    v8f acc0 = {}, acc1 = {}, acc2 = {}, acc3 = {};
    const float* arow = E + mRow * EMB_DIM + kb;

    for (int k = 0; k < EMB_DIM; k += 4) {
        v2f a = *(const v2f*)(arow + k);                    // A[M][k+kb .. k+kb+1]
        const float* b0p = Wf + (k + kb) * PROJ_DIM + nCol; // B[k+kb][N]
        const float* b1p = b0p + PROJ_DIM;                  // B[k+kb+1][N]
        v2f b0 = { b0p[0],  b1p[0]  };
        v2f b1 = { b0p[16], b1p[16] };
        v2f b2 = { b0p[32], b1p[32] };
        v2f b3 = { b0p[48], b1p[48] };
        acc0 = __builtin_amdgcn_wmma_f32_16x16x4_f32(false, a, false, b0, (short)0, acc0, false, false);
        acc1 = __builtin_amdgcn_wmma_f32_16x16x4_f32(false, a, false, b1, (short)0, acc1, false, false);
        acc2 = __builtin_amdgcn_wmma_f32_16x16x4_f32(false, a, false, b2, (short)0, acc2, false, false);
        acc3 = __builtin_amdgcn_wmma_f32_16x16x4_f32(false, a, false, b3, (short)0, acc3, false, false);
    }

    const int mOut = mBase + ((lane >> 4) << 3);  // +0 or +8
    float* xcol = X + nBase + (lane & 15);
    #pragma unroll
    for (int v = 0; v < 8; ++v) {
        float* p = xcol + (mOut + v) * PROJ_DIM;
        p[0]  = acc0[v];
        p[16] = acc1[v];
        p[32] = acc2[v];
        p[48] = acc3[v];
    }
}

// ---------------------------------------------------------------- kernel 4
// a_src[n,h] = x[n,h,:].att_src[h,:],  a_dst likewise (one wave per node)
__global__ void __launch_bounds__(256) ga_attn_coef_kernel(const float* __restrict__ X,
                                                           const float* __restrict__ att_src,
                                                           const float* __restrict__ att_dst,
                                                           float* __restrict__ a_src,
                                                           float* __restrict__ a_dst) {
    const int n    = blockIdx.x * 8 + (threadIdx.x >> 5);
    const int lane = threadIdx.x & 31;
    const float* x = X + n * PROJ_DIM;
    #pragma unroll
    for (int h = 0; h < HEADS; ++h) {
        const float* xe = x + h * EMB_DIM;
        const float* as = att_src + h * EMB_DIM;
        const float* ad = att_dst + h * EMB_DIM;
        float ss = 0.f, sd = 0.f;
        for (int k = lane; k < EMB_DIM; k += 32) {
            float v = xe[k];
            ss += v * as[k];
            sd += v * ad[k];
        }
        ss = wave_sum(ss);
        sd = wave_sum(sd);
        if (lane == 0) { a_src[n * HEADS + h] = ss; a_dst[n * HEADS + h] = sd; }
    }
}

// ---------------------------------------------------------------- kernel 5
// Per destination i: banded leaky-relu softmax (23 x 4 in LDS) + aggregation,
// out[i,t] = mean_h sum_j alpha[i,j,h] x[j,h,t] + bias[t]
__global__ void __launch_bounds__(384) ga_aggregate_kernel(const float* __restrict__ X,
                                                           const float* __restrict__ a_src,
                                                           const float* __restrict__ a_dst,
                                                           const unsigned char* __restrict__ mask,
                                                           const float* __restrict__ bias,
                                                           float* __restrict__ out) {
    __shared__ float alpha[BANDW][HEADS];
    const int i = blockIdx.x;
    const int t = threadIdx.x;  // 0..383

    if (t < HEADS) {
        const int h = t;
        const float ad = a_dst[i * HEADS + h];
        float lg[BANDW];
        bool  vl[BANDW];
        float mx = -3.0e38f;
        #pragma unroll
        for (int jj = 0; jj < BANDW; ++jj) {
            const int j = i - BAND + jj;
            const bool v = (j >= 0) && (j < N_SPANS) && mask[i * BANDW + jj];
            vl[jj] = v;
            float l = 0.f;
            if (v) {
                l = ad + a_src[j * HEADS + h];
                l = (l > 0.f) ? l : NEG_SLOPE * l;
                mx = fmaxf(mx, l);
            }
            lg[jj] = l;
        }
        float sum = 0.f;
        #pragma unroll
        for (int jj = 0; jj < BANDW; ++jj) {
            const float e = vl[jj] ? __expf(lg[jj] - mx) : 0.f;
            lg[jj] = e;
            sum += e;
        }
        const float inv = 1.f / sum;   // self-loop guarantees sum > 0
        #pragma unroll
        for (int jj = 0; jj < BANDW; ++jj) alpha[jj][h] = lg[jj] * inv;
    }
    __syncthreads();

    float acc = 0.f;
    for (int jj = 0; jj < BANDW; ++jj) {
        const int j = i - BAND + jj;
        if (j < 0 || j >= N_SPANS) continue;
        const float a0 = alpha[jj][0], a1 = alpha[jj][1],
                    a2 = alpha[jj][2], a3 = alpha[jj][3];
        if (a0 == 0.f && a1 == 0.f && a2 == 0.f && a3 == 0.f) continue;
        const float* xj = X + j * PROJ_DIM + t;   // coalesced over t
        acc += a0 * xj[0] + a1 * xj[EMB_DIM] + a2 * xj[2 * EMB_DIM] + a3 * xj[3 * EMB_DIM];
    }
    out[i * EMB_DIM + t] = acc * (1.f / HEADS) + bias[t];
}

// ---------------------------------------------------------------- launch
extern "C" void kernel_launch(void* const* d_in, const int* in_sizes, int n_in,
                              void* d_out, int out_size, void* d_ws, size_t ws_size,
                              hipStream_t stream) {
    const float* E        = (const float*)d_in[0];
    // d_in[1] = span_positions (int64) — unused by the reference output
    const float* W        = (const float*)d_in[2];  // [384][4][384] row-major == [384][1536]
    const float* att_src  = (const float*)d_in[3];
    const float* att_dst  = (const float*)d_in[4];
    const float* bias     = (const float*)d_in[5];
    float* out            = (float*)d_out;

    char* ws = (char*)d_ws;
    float* X       = (float*)ws;  ws += (size_t)N_SPANS * PROJ_DIM * sizeof(float); // 25.2 MB
    float* rnorm   = (float*)ws;  ws += (size_t)N_SPANS * sizeof(float);
    float* a_src   = (float*)ws;  ws += (size_t)N_SPANS * HEADS * sizeof(float);
    float* a_dst   = (float*)ws;  ws += (size_t)N_SPANS * HEADS * sizeof(float);
    unsigned char* mask = (unsigned char*)ws;                                       // 4096*23 B

    ga_norm_kernel<<<N_SPANS / 8, 256, 0, stream>>>(E, rnorm);
    ga_band_mask_kernel<<<N_SPANS, 256, 0, stream>>>(E, rnorm, mask);
    ga_gemm_wmma_kernel<<<dim3(PROJ_DIM / 64, N_SPANS / (16 * 8)), 256, 0, stream>>>(E, W, X);
    ga_attn_coef_kernel<<<N_SPANS / 8, 256, 0, stream>>>(X, att_src, att_dst, a_src, a_dst);
    ga_aggregate_kernel<<<N_SPANS, 384, 0, stream>>>(X, a_src, a_dst, mask, bias, out);
}